// InvertAffine_34110630265651
// MI455X (gfx1250) — compile-verified
//
#include <hip/hip_runtime.h>
#include <stdint.h>

// InvertAffine: batched inverse of (I + [E t; 0 0]) 4x4 affines, returned as
// flattened shift (inv - I, top 3 rows). Pure streaming problem:
// 192 MB @ 23.3 TB/s => ~8.2 us floor. Bandwidth path: CDNA5 async
// global<->LDS b128 copies (coalesced 512B/wave) + padded LDS staging.

namespace {
constexpr int kBlock   = 256;  // samples per block (1 per thread)
constexpr int kStrideF = 14;   // floats per LDS slot: 12 used + 2 pad (56B)
}

__device__ __forceinline__ uint32_t lds_off_u32(const void* p) {
  // For __shared__ flat pointers the group-segment byte offset is bits [31:0]
  // of the flat address (LDS aperture lives in the high bits).
  return (uint32_t)(uintptr_t)p;
}

__device__ __forceinline__ void async_g2lds_b128(uint32_t ldsoff, const float* g) {
  // ASYNCcnt-tracked DMA of 16B/lane straight into LDS (no VGPR staging).
  asm volatile("global_load_async_to_lds_b128 %0, %1, off"
               :: "v"(ldsoff), "v"(g)
               : "memory");
}

__device__ __forceinline__ void async_lds2g_b128(float* g, uint32_t ldsoff) {
  asm volatile("global_store_async_from_lds_b128 %0, %1, off"
               :: "v"(g), "v"(ldsoff)
               : "memory");
}

__device__ __forceinline__ void wait_async0() {
  asm volatile("s_wait_asynccnt 0x0" ::: "memory");
}

__global__ __launch_bounds__(kBlock)
void InvertAffine_34110630265651_kernel(const float* __restrict__ trf,
                                        float* __restrict__ out,
                                        int nsamp) {
  __shared__ float lds[kBlock * kStrideF];

  const int t          = threadIdx.x;
  const int blockStart = blockIdx.x * kBlock;
  const bool fullBlock = (blockStart + kBlock) <= nsamp;   // uniform per block
  const bool active    = (blockStart + t) < nsamp;

  float m[12];

  if (fullBlock) {
    // --- coalesced async load: 768 contiguous float4 chunks -> padded LDS ---
    const float* gbase = trf + (size_t)blockStart * 12;
#pragma unroll
    for (int k = 0; k < 3; ++k) {
      const int chunk = t + k * kBlock;        // lanes hit consecutive chunks
      const int s  = chunk / 3;
      const int sl = chunk - 3 * s;
      async_g2lds_b128(lds_off_u32(&lds[s * kStrideF + sl * 4]),
                       gbase + (size_t)chunk * 4);
    }
    wait_async0();      // this wave's async loads complete
    __syncthreads();    // all waves' loads visible block-wide
    const float* slot = &lds[t * kStrideF];
#pragma unroll
    for (int i = 0; i < 12; ++i) m[i] = slot[i];
  } else if (active) {
    // tail block: plain vectorized path, no barriers needed
    const float4* g = reinterpret_cast<const float4*>(trf) +
                      (size_t)(blockStart + t) * 3;
    float4 r0 = g[0], r1 = g[1], r2 = g[2];
    m[0] = r0.x; m[1] = r0.y; m[2]  = r0.z; m[3]  = r0.w;
    m[4] = r1.x; m[5] = r1.y; m[6]  = r1.z; m[7]  = r1.w;
    m[8] = r2.x; m[9] = r2.y; m[10] = r2.z; m[11] = r2.w;
  }

  float o[12];
  if (active) {
    // full = I + shift; A = 3x3 block, tvec = last column
    const float a00 = m[0] + 1.0f, a01 = m[1],        a02 = m[2],         t0 = m[3];
    const float a10 = m[4],        a11 = m[5] + 1.0f, a12 = m[6],         t1 = m[7];
    const float a20 = m[8],        a21 = m[9],        a22 = m[10] + 1.0f, t2 = m[11];

    // cofactor (adjugate) inverse of A
    const float c00 = a11 * a22 - a12 * a21;
    const float c01 = a12 * a20 - a10 * a22;
    const float c02 = a10 * a21 - a11 * a20;
    const float det = a00 * c00 + a01 * c01 + a02 * c02;
    const float r   = 1.0f / det;

    const float i00 = c00 * r;
    const float i10 = c01 * r;
    const float i20 = c02 * r;
    const float i01 = (a02 * a21 - a01 * a22) * r;
    const float i11 = (a00 * a22 - a02 * a20) * r;
    const float i21 = (a01 * a20 - a00 * a21) * r;
    const float i02 = (a01 * a12 - a02 * a11) * r;
    const float i12 = (a02 * a10 - a00 * a12) * r;
    const float i22 = (a00 * a11 - a01 * a10) * r;

    // translation of inverse: -A^{-1} t
    const float o03 = -(i00 * t0 + i01 * t1 + i02 * t2);
    const float o13 = -(i10 * t0 + i11 * t1 + i12 * t2);
    const float o23 = -(i20 * t0 + i21 * t1 + i22 * t2);

    // shift = inv - I (top 3 rows, flattened)
    o[0] = i00 - 1.0f; o[1] = i01;        o[2]  = i02;        o[3]  = o03;
    o[4] = i10;        o[5] = i11 - 1.0f; o[6]  = i12;        o[7]  = o13;
    o[8] = i20;        o[9] = i21;        o[10] = i22 - 1.0f; o[11] = o23;
  }

  if (fullBlock) {
    // --- results back to own padded slot, then coalesced async store ---
    float* slot = &lds[t * kStrideF];
#pragma unroll
    for (int i = 0; i < 12; ++i) slot[i] = o[i];
    __syncthreads();    // all slots written before DMA reads LDS
    float* obase = out + (size_t)blockStart * 12;
#pragma unroll
    for (int k = 0; k < 3; ++k) {
      const int chunk = t + k * kBlock;
      const int s  = chunk / 3;
      const int sl = chunk - 3 * s;
      async_lds2g_b128(obase + (size_t)chunk * 4,
                       lds_off_u32(&lds[s * kStrideF + sl * 4]));
    }
    wait_async0();      // drain before wave retires
  } else if (active) {
    float4* g = reinterpret_cast<float4*>(out) + (size_t)(blockStart + t) * 3;
    g[0] = make_float4(o[0], o[1], o[2],  o[3]);
    g[1] = make_float4(o[4], o[5], o[6],  o[7]);
    g[2] = make_float4(o[8], o[9], o[10], o[11]);
  }
}

extern "C" void kernel_launch(void* const* d_in, const int* in_sizes, int n_in,
                              void* d_out, int out_size, void* d_ws, size_t ws_size,
                              hipStream_t stream) {
  (void)n_in; (void)out_size; (void)d_ws; (void)ws_size;
  const float* trf = (const float*)d_in[0];
  float* out       = (float*)d_out;
  const int nsamp  = in_sizes[0] / 12;          // B = 2,000,000
  const int blocks = (nsamp + kBlock - 1) / kBlock;
  InvertAffine_34110630265651_kernel<<<dim3(blocks), dim3(kBlock), 0, stream>>>(
      trf, out, nsamp);
}